// OuterProductNetworkLayer_25872882991601
// MI455X (gfx1250) — compile-verified
//
#include <hip/hip_runtime.h>

#define B_ 1024
#define N_ 40
#define E_ 64
#define P_ 780
#define WT_STRIDE 66   // padded row stride for transposed W in LDS

typedef float v2f __attribute__((ext_vector_type(2)));
typedef float v4f __attribute__((ext_vector_type(4)));
typedef float v8f __attribute__((ext_vector_type(8)));
typedef int   v4i __attribute__((vector_size(4 * sizeof(int))));

#define AS1 __attribute__((address_space(1)))
#define AS3 __attribute__((address_space(3)))

#if defined(__has_builtin)
#if __has_builtin(__builtin_amdgcn_global_load_async_to_lds_b128) && \
    __has_builtin(__builtin_amdgcn_global_load_async_to_lds_b32)
#define HAS_ASYNC 1
#endif
#endif
#ifndef HAS_ASYNC
#define HAS_ASYNC 0
#endif

#define WAIT_ASYNC_8() asm volatile("s_wait_asynccnt 0x8" ::: "memory")
#define WAIT_ASYNC_0() asm volatile("s_wait_asynccnt 0x0" ::: "memory")

// One block per pair i: T = Q_i (16b x 64f) * W_i (64f x 64e) via
// V_WMMA_F32_16X16X4_F32, then out[b,i] = sum_e P_i[b,e] * T[b,e].
__global__ __launch_bounds__(256) void opn_pair_kernel(
    const float* __restrict__ inputs,   // [B, N, E]
    const float* __restrict__ param,    // [E(f), P, E(e)]
    float* __restrict__ out)            // [B, 1, P] flat = b*P + i
{
  __shared__ float WT[E_ * WT_STRIDE];     // W^T: WT[e*66 + f] = param[f,pair,e]
  __shared__ float Qt[8][2][16 * E_];      // per-wave double-buffered Q tile

  const int pair = blockIdx.x;

  // derive (row, col) of this pair from the upper-triangular enumeration
  int t = pair, r = 0, rem = N_ - 1;
  while (t >= rem) { t -= rem; ++r; --rem; }
  const int row = r;
  const int col = r + 1 + t;

  const int tid  = threadIdx.x;
  const int wave = tid >> 5;
  const int lane = tid & 31;
  const int half = lane >> 4;   // 0: lanes 0-15, 1: lanes 16-31
  const int l16  = lane & 15;

  const float* wsrc = param + (size_t)pair * E_;

#if HAS_ASYNC
  // ---- stage W_i^T via async b32 (global contiguous along e, LDS transposed)
  {
    #pragma unroll
    for (int it = 0; it < 16; ++it) {
      int g = it * 256 + tid;            // element id 0..4095
      int e = g & 63, f = g >> 6;
      __builtin_amdgcn_global_load_async_to_lds_b32(
          (AS1 int*)(wsrc + (size_t)f * (P_ * E_) + e),
          (AS3 int*)(&WT[e * WT_STRIDE + f]), 0, 0);
    }
  }
  // ---- async Q-tile stage: 8 x b128 per wave into chosen buffer
  auto stage_q = [&](int bt_, float* buf) {
    const int m0_ = bt_ << 4;
    #pragma unroll
    for (int it = 0; it < 8; ++it) {
      int c  = it * 32 + lane;           // 0..255 float4 chunks
      int m  = c >> 4;
      int e4 = (c & 15) << 2;
      __builtin_amdgcn_global_load_async_to_lds_b128(
          (AS1 v4i*)(inputs + (size_t)(m0_ + m) * (N_ * E_) +
                     (size_t)col * E_ + e4),
          (AS3 v4i*)(buf + m * E_ + e4), 0, 0);
    }
  };
  // prologue: first Q tile behind W; in-order ASYNCcnt: wait<=8 => W(16) done
  stage_q(wave, &Qt[wave][0][0]);
  WAIT_ASYNC_8();
  __syncthreads();   // WT visible to all waves
#else
  // ---- fallback: sync W transpose stage
  {
    #pragma unroll
    for (int it = 0; it < 4; ++it) {
      int c  = it * 256 + tid;
      int f  = c >> 4;
      int e4 = (c & 15) << 2;
      v4f w = *(const v4f*)(wsrc + (size_t)f * (P_ * E_) + e4);
      WT[(e4 + 0) * WT_STRIDE + f] = w.x;
      WT[(e4 + 1) * WT_STRIDE + f] = w.y;
      WT[(e4 + 2) * WT_STRIDE + f] = w.z;
      WT[(e4 + 3) * WT_STRIDE + f] = w.w;
    }
  }
  __syncthreads();
#endif

  int cur = 0;
  // 64 batch tiles of 16 rows; 8 tiles per wave (uniform trip count)
  for (int bt = wave; bt < B_ / 16; bt += 8) {
    const int m0 = bt << 4;
    float* qt = &Qt[wave][cur][0];

#if HAS_ASYNC
    const int next = bt + 8;
    if (next < B_ / 16) {
      stage_q(next, &Qt[wave][cur ^ 1][0]);
      WAIT_ASYNC_8();   // 8+8 outstanding -> tile bt resident (in-order)
    } else {
      WAIT_ASYNC_0();
    }
#else
    #pragma unroll
    for (int it = 0; it < 8; ++it) {
      int c  = it * 32 + lane;
      int m  = c >> 4;
      int e4 = (c & 15) << 2;
      v4f q = *(const v4f*)(inputs + (size_t)(m0 + m) * (N_ * E_) +
                            (size_t)col * E_ + e4);
      *(v4f*)(qt + m * E_ + e4) = q;
    }
    asm volatile("s_wait_dscnt 0" ::: "memory");
#endif

    // ---- A fragments: 16x4 fp32 layout (lanes 0-15: K,K+1; 16-31: K+2,K+3)
    v2f aF[16];
    #pragma unroll
    for (int kk = 0; kk < 16; ++kk)
      aF[kk] = *(const v2f*)(qt + l16 * E_ + kk * 4 + half * 2);

    float partial[8];
    #pragma unroll
    for (int rr = 0; rr < 8; ++rr) partial[rr] = 0.f;

    #pragma unroll
    for (int nt = 0; nt < 4; ++nt) {
      const int n = nt * 16 + l16;
      v8f d = {0.f, 0.f, 0.f, 0.f, 0.f, 0.f, 0.f, 0.f};
      #pragma unroll
      for (int kk = 0; kk < 16; ++kk) {
        // B fragment: (W[k,n], W[k+1,n]) = contiguous pair in transposed LDS
        v2f bF = *(const v2f*)(&WT[n * WT_STRIDE + kk * 4 + half * 2]);
        d = __builtin_amdgcn_wmma_f32_16x16x4_f32(
                /*neg_a=*/false, aF[kk], /*neg_b=*/false, bF,
                /*c_mod=*/(short)0, d, /*reuse_a=*/false, /*reuse_b=*/false);
      }
      // D layout: VGPR rr -> M = rr + 8*half, N = l16 (+ nt*16)
      #pragma unroll
      for (int rr = 0; rr < 8; ++rr) {
        int brow = m0 + rr + half * 8;
        float pv = inputs[(size_t)brow * (N_ * E_) + (size_t)row * E_ +
                          nt * 16 + l16];
        partial[rr] += pv * d[rr];
      }
    }

    // ---- butterfly-reduce the 16 lanes of each half, then write out[b*P+i]
    #pragma unroll
    for (int rr = 0; rr < 8; ++rr) {
      float v = partial[rr];
      v += __shfl_xor(v, 1, 32);
      v += __shfl_xor(v, 2, 32);
      v += __shfl_xor(v, 4, 32);
      v += __shfl_xor(v, 8, 32);
      if (l16 == 0) {
        int brow = m0 + rr + half * 8;
        out[(size_t)brow * P_ + pair] = v;
      }
    }
    cur ^= 1;
  }
}

extern "C" void kernel_launch(void* const* d_in, const int* in_sizes, int n_in,
                              void* d_out, int out_size, void* d_ws, size_t ws_size,
                              hipStream_t stream) {
  const float* inputs = (const float*)d_in[0];  // B*N*E fp32
  const float* param  = (const float*)d_in[1];  // E*P*E fp32
  float* out = (float*)d_out;                   // B*P fp32
  opn_pair_kernel<<<P_, 256, 0, stream>>>(inputs, param, out);
}